// SBS1dLayer_55533927137911
// MI455X (gfx1250) — compile-verified
//
#include <hip/hip_runtime.h>
#include <hip/hip_bf16.h>

// MPS classifier forward for MI455X (gfx1250).
// Scan kernel: 8 blocks = (side in {L,R}) x (string s). 256 threads = 8 wave32.
// Per site: V'(64x32) = W(64x64) x A_site(64x32) via v_wmma_f32_16x16x4_f32.
// Site weights streamed into LDS with global_load_async_to_lds_b128 (ASYNCcnt,
// inline asm); state V and weights both double-buffered -> one barrier per site.
// Right-side transpose is folded into the B-fragment LDS addressing.

#define S_     4
#define B_     64
#define D_     32
#define N_     784
#define HALF_  392
#define L_     10
#define NSITES 391      // transfer sites per side
#define LDA    44       // LDS row stride for A_site (floats); 176B row, 16B aligned
#define LDV    33       // LDS row stride for state V (floats); conflict-free A-frag gather

typedef __attribute__((ext_vector_type(2))) float v2f;
typedef __attribute__((ext_vector_type(8))) float v8f;

// async copy of one 16B chunk: global -> LDS, tracked by ASYNCcnt.
// lds address = low 32 bits of the flat pointer (AS3->flat cast keeps the
// LDS offset in the low word); global address = 64-bit VGPR pair.
static __device__ __forceinline__ void async_g2l_b128(void* ldsDst, const void* gSrc) {
    const uint32_t laddr = (uint32_t)(uintptr_t)ldsDst;
    const uint64_t gaddr = (uint64_t)(uintptr_t)gSrc;
    asm volatile("global_load_async_to_lds_b128 %0, %1, off"
                 :: "v"(laddr), "v"(gaddr) : "memory");
}
static __device__ __forceinline__ void wait_async0() {
#if __has_builtin(__builtin_amdgcn_s_wait_asynccnt)
    __builtin_amdgcn_s_wait_asynccnt(0);
#else
    asm volatile("s_wait_asynccnt 0x0" ::: "memory");
#endif
}

// Branch-free K=64 WMMA chain: 16 steps of v_wmma_f32_16x16x4_f32.
// A-frag (16x4 f32): lanes 0-15 hold K=k,k+1; lanes 16-31 K=k+2,k+3;
// built as W[b, 2u+p] = V[b,u]*x[b,p]. B-frag strides are compile-time:
//   left  (BSTEP=4*LDA): B[c=i*2+p][j]    -> row c,          col n0+l
//   right (BSTEP=2)    : B[c=j*2+p][i]    -> row (n0+l)*2+p, col j
template <int BSTEP>
static __device__ __forceinline__ v8f kloop64(const float* __restrict__ vbuf,
                                              const float* __restrict__ abuf,
                                              int vaddr, int baddr,
                                              float x0, float x1) {
    v8f c = {};
    #pragma unroll
    for (int kk = 0; kk < 16; ++kk) {
        const float vv = vbuf[vaddr + kk * 2];
        v2f a = { vv * x0, vv * x1 };
        v2f b = { abuf[baddr + kk * BSTEP], abuf[baddr + kk * BSTEP + LDA] };
        c = __builtin_amdgcn_wmma_f32_16x16x4_f32(
                false, a, false, b, (short)0, c, false, false);
    }
    return c;
}

__global__ __launch_bounds__(256) void sbs_scan_kernel(
    const float* __restrict__ x,        // (B, N, P)
    const float* __restrict__ A_first,  // (S, D, P)
    const float* __restrict__ A_left,   // (S, 391, D, P, D)
    const float* __restrict__ A_right,  // (S, 391, D, P, D)
    const float* __restrict__ A_last,   // (S, D, P)
    float* __restrict__ out_v,          // (S, B, D)
    float* __restrict__ out_r)          // (S, B, D)
{
    __shared__ float ldsV[2][B_ * LDV];                   // state, dbl-buffered
    __shared__ __align__(16) float ldsA[2][64 * LDA];     // site tile, dbl-buffered

    const int tid  = threadIdx.x;
    const int side = blockIdx.x & 1;          // 0 = left scan, 1 = right scan
    const int s    = blockIdx.x >> 1;

    const int SITE_ELTS = D_ * 2 * D_;        // 2048 floats per site
    const float* Abase = (side ? A_right : A_left) + (size_t)s * NSITES * SITE_ELTS;
    const ptrdiff_t dA = side ? -(ptrdiff_t)SITE_ELTS : (ptrdiff_t)SITE_ELTS;
    const float* pA = side ? (Abase + (size_t)(NSITES - 1) * SITE_ELTS) : Abase;

    // issue this thread's share (2 x b128) of one 8KB site tile into LDS,
    // preserving memory order [(i*2+p)*LDA + j] with row padding.
    auto issueSite = [&](const float* src, float* dst) {
        #pragma unroll
        for (int u = 0; u < 2; ++u) {
            const int e   = tid + u * 256;         // b128 chunk id (0..511)
            const int row = e >> 3;                // (i*2+p)
            const int col = (e & 7) * 4;           // j (multiple of 4)
            async_g2l_b128(&dst[row * LDA + col], src + e * 4);
        }
    };

    // ---- stage site 0 + initialize state from the boundary tensor ----
    issueSite(pA, ldsA[0]);
    {
        const float* A0 = (side ? A_last : A_first) + s * D_ * 2;  // (D, P)
        const int x_site = side ? (N_ - 1) : 0;
        for (int e = tid; e < B_ * D_; e += 256) {
            const int b = e >> 5, i = e & 31;
            const float2 xv = *(const float2*)&x[((size_t)b * N_ + x_site) * 2];
            ldsV[0][b * LDV + i] = A0[i * 2 + 0] * xv.x + A0[i * 2 + 1] * xv.y;
        }
    }
    wait_async0();
    __syncthreads();

    const int lane = tid & 31;                // wave32
    const int wave = tid >> 5;                // 0..7
    const int l    = lane & 15;
    const int half = lane >> 4;
    const int m0   = (wave & 3) * 16;         // batch-row tile origin (M)
    const int n0   = (wave >> 2) * 16;        // output-col tile origin (N)
    const int brow = m0 + l;                  // this lane's batch row

    // per-lane running input pointer (x[brow, site, :])
    const int xs0  = side ? (HALF_ + NSITES - 1) : 1;
    const int dxs  = side ? -2 : 2;           // floats per site step
    const float* px = x + ((size_t)brow * N_ + xs0) * 2;

    const int vaddr = brow * LDV + half;
    const int baddrL = 2 * half * LDA + n0 + l;
    const int baddrR = (n0 + l) * 2 * LDA + half;

    int cur = 0;
    for (int t = 0; t < NSITES; ++t) {
        const float* ldA  = ldsA[cur];
        const float* ldVc = ldsV[cur];
        float*       ldVn = ldsV[cur ^ 1];

        // start streaming next site into the other buffer (overlaps WMMA loop)
        if (t + 1 < NSITES) issueSite(pA + dA, ldsA[cur ^ 1]);
        if (t + 2 < NSITES)
            __builtin_prefetch(pA + 2 * dA + tid * 8, 0, 1);   // global_prefetch_b8

        const float2 xv = *(const float2*)px;  px += dxs;

        const v8f c = (side == 0)
            ? kloop64<4 * LDA>(ldVc, ldA, vaddr, baddrL, xv.x, xv.y)
            : kloop64<2>(ldVc, ldA, vaddr, baddrR, xv.x, xv.y);

        // write V' tile into the other state buffer (C/D: VGPR r -> M = r + 8*half)
        #pragma unroll
        for (int r8 = 0; r8 < 8; ++r8)
            ldVn[(m0 + r8 + 8 * half) * LDV + n0 + l] = c[r8];

        wait_async0();     // next site's tile landed in LDS (no-op on last iter)
        __syncthreads();   // publishes V' and the new A tile; seals this site's reads
        pA += dA;
        cur ^= 1;
    }

    // emit final state for this (side, s); last write went to ldsV[NSITES & 1]
    const float* fin = ldsV[NSITES & 1];
    float* out = (side ? out_r : out_v) + (size_t)s * B_ * D_;
    for (int e = tid; e < B_ * D_; e += 256) {
        const int b = e >> 5, i = e & 31;
        out[b * D_ + i] = fin[b * LDV + i];
    }
}

// out[b,l] = prod_s ( v_s^T . A_label[s,:,l,:] . r_s )
__global__ __launch_bounds__(256) void sbs_combine_kernel(
    const float* __restrict__ A_label,  // (S, D, L, D)
    const float* __restrict__ ws_v,     // (S, B, D)
    const float* __restrict__ ws_r,     // (S, B, D)
    float* __restrict__ out)            // (B, L)
{
    const int idx = blockIdx.x * 256 + threadIdx.x;
    if (idx >= B_ * L_) return;
    const int b = idx / L_, lb = idx % L_;

    float prod = 1.0f;
    for (int s = 0; s < S_; ++s) {
        const float* v = ws_v + ((size_t)s * B_ + b) * D_;
        const float* r = ws_r + ((size_t)s * B_ + b) * D_;
        float acc = 0.0f;
        for (int d = 0; d < D_; ++d) {
            const float* Arow = A_label + (((size_t)s * D_ + d) * L_ + lb) * D_;
            float dot = 0.0f;
            #pragma unroll
            for (int j = 0; j < D_; ++j) dot = fmaf(Arow[j], r[j], dot);
            acc = fmaf(v[d], dot, acc);
        }
        prod *= acc;
    }
    out[idx] = prod;
}

extern "C" void kernel_launch(void* const* d_in, const int* in_sizes, int n_in,
                              void* d_out, int out_size, void* d_ws, size_t ws_size,
                              hipStream_t stream) {
    const float* x       = (const float*)d_in[0];
    const float* A_first = (const float*)d_in[1];
    const float* A_left  = (const float*)d_in[2];
    const float* A_label = (const float*)d_in[3];
    const float* A_right = (const float*)d_in[4];
    const float* A_last  = (const float*)d_in[5];

    float* ws_v = (float*)d_ws;                 // S*B*D f32
    float* ws_r = ws_v + S_ * B_ * D_;          // S*B*D f32

    sbs_scan_kernel<<<2 * S_, 256, 0, stream>>>(
        x, A_first, A_left, A_right, A_last, ws_v, ws_r);
    sbs_combine_kernel<<<(B_ * L_ + 255) / 256, 256, 0, stream>>>(
        A_label, ws_v, ws_r, (float*)d_out);
}